// LRU_36000415875649
// MI455X (gfx1250) — compile-verified
//
#include <hip/hip_runtime.h>

#define IN_F    128
#define OUT_F   128
#define STATE_F 256
#define BATCH   8
#define TLEN    8192
#define MTOT    (BATCH * TLEN)     /* 65536 rows of (b,t) */
#define CHUNK   256
#define NCHUNK  (TLEN / CHUNK)     /* 32 */

#define PADB 132   /* 128 + 4 : stride%64 = 4 -> conflict-free row access */
#define PADC 260   /* 256 + 4 */

typedef __attribute__((ext_vector_type(2))) float v2f;
typedef __attribute__((ext_vector_type(8))) float v8f;

__device__ __forceinline__ v8f wmma_f32_k4(v2f a, v2f b, v8f c) {
  // D(16x16,f32) = A(16x4,f32) * B(4x16,f32) + C
  return __builtin_amdgcn_wmma_f32_16x16x4_f32(
      /*neg_a=*/false, a, /*neg_b=*/false, b,
      /*c_mod=*/(short)0, c, /*reuse_a=*/false, /*reuse_b=*/false);
}

__device__ __forceinline__ void lambda_of(int n, const float* nu_log,
                                          const float* theta_log,
                                          float& lre, float& lim) {
  float lm = expf(-expf(nu_log[n]));
  float th = expf(theta_log[n]);
  lre = lm * cosf(th);
  lim = lm * sinf(th);
}

// ---------------------------------------------------------------------------
// Kernel 1: Bu_{re,im} = x @ (gamma*B_{re,im})^T.
// WG = (ntile, 8 consecutive mtiles). gamma*B rows staged once in LDS and
// shared by all 8 waves; each wave reuses its A fragment for re+im WMMAs.
// ---------------------------------------------------------------------------
__global__ __launch_bounds__(256) void k_bu_gemm(
    const float* __restrict__ x, const float* __restrict__ Bre,
    const float* __restrict__ Bim, const float* __restrict__ gamma_log,
    float* __restrict__ bu_re, float* __restrict__ bu_im) {
  __shared__ float sB[2][16][PADB];
  const int tid   = threadIdx.x;
  const int ntile = blockIdx.x & 15;       // STATE_F/16 = 16
  const int mgrp  = blockIdx.x >> 4;       // 512 groups of 8 mtiles
  const int n0    = ntile << 4;

  // stage gamma-scaled B_re|B_im tiles: 2*16*128 floats = 1024 float4
  for (int q = tid; q < 1024; q += 256) {
    const int part = q >> 9;               // 512 float4 per part
    const int r    = (q >> 5) & 15;        // 32 float4 per row
    const int kq   = (q & 31) << 2;
    const float g  = expf(gamma_log[n0 + r]);
    const float4 v = *(const float4*)((part ? Bim : Bre) +
                                      (size_t)(n0 + r) * IN_F + kq);
    float* dst = &sB[part][r][kq];
    dst[0] = v.x * g; dst[1] = v.y * g; dst[2] = v.z * g; dst[3] = v.w * g;
  }
  __syncthreads();

  const int lane = tid & 31, wave = tid >> 5;
  const int m0   = (mgrp * 8 + wave) << 4;
  const int row  = lane & 15, khalf = lane >> 4;

  const float* xrow = x + (size_t)(m0 + row) * IN_F + khalf * 2;
  const float* brre = &sB[0][row][khalf * 2];
  const float* brim = &sB[1][row][khalf * 2];

  v8f accR = {}, accI = {};
#pragma unroll
  for (int k0 = 0; k0 < IN_F; k0 += 4) {
    v2f a  = *(const v2f*)(xrow + k0);     // one A fragment, two WMMAs
    v2f br = *(const v2f*)(brre + k0);     // ds_load_b64
    v2f bi = *(const v2f*)(brim + k0);
    accR = wmma_f32_k4(a, br, accR);
    accI = wmma_f32_k4(a, bi, accI);
  }

  float* oR = bu_re + n0 + row;
  float* oI = bu_im + n0 + row;
#pragma unroll
  for (int r = 0; r < 8; ++r) {
    const size_t off = (size_t)(m0 + r + khalf * 8) * STATE_F;
    oR[off] = accR[r];
    oI[off] = accI[r];
  }
}

// ---------------------------------------------------------------------------
// Kernel 2a: in-place local scan (zero init) within each chunk of 256 steps.
// One thread per (b, chunk, n). Coalesced over n. Emits chunk-end values.
// ---------------------------------------------------------------------------
__global__ __launch_bounds__(256) void k_scan_local(
    float* __restrict__ bu_re, float* __restrict__ bu_im,
    const float* __restrict__ nu_log, const float* __restrict__ theta_log,
    float* __restrict__ el_re, float* __restrict__ el_im) {
  const int c = blockIdx.x & (NCHUNK - 1);
  const int b = blockIdx.x >> 5;
  const int n = threadIdx.x;
  float lre, lim;
  lambda_of(n, nu_log, theta_log, lre, lim);

  size_t base = ((size_t)b * TLEN + (size_t)c * CHUNK) * STATE_F + n;
  float sre = 0.f, sim = 0.f;
  for (int j = 0; j < CHUNK; ++j) {
    size_t idx = base + (size_t)j * STATE_F;
    if (j + 16 < CHUNK) {
      __builtin_prefetch(&bu_re[idx + 16 * STATE_F], 0, 1);
      __builtin_prefetch(&bu_im[idx + 16 * STATE_F], 0, 1);
    }
    float ure = bu_re[idx], uim = bu_im[idx];
    float nre = fmaf(lre, sre, fmaf(-lim, sim, ure));
    float nim = fmaf(lre, sim, fmaf(lim, sre, uim));
    sre = nre; sim = nim;
    bu_re[idx] = sre; bu_im[idx] = sim;
  }
  int ei = (b * NCHUNK + c) * STATE_F + n;
  el_re[ei] = sre; el_im[ei] = sim;
}

// ---------------------------------------------------------------------------
// Kernel 2b: carry scan over chunks. carry[c] = state entering chunk c.
// ---------------------------------------------------------------------------
__global__ __launch_bounds__(256) void k_carry(
    const float* __restrict__ nu_log, const float* __restrict__ theta_log,
    const float* __restrict__ el_re, const float* __restrict__ el_im,
    float* __restrict__ cr_re, float* __restrict__ cr_im) {
  const int idx = blockIdx.x * 256 + threadIdx.x;
  const int b = idx >> 8, n = idx & 255;
  float lre, lim;
  lambda_of(n, nu_log, theta_log, lre, lim);
  // Lambda^CHUNK via 8 complex squarings (2^8 = 256)
  float pre = lre, pim = lim;
#pragma unroll
  for (int s = 0; s < 8; ++s) {
    float qre = pre * pre - pim * pim;
    float qim = 2.f * pre * pim;
    pre = qre; pim = qim;
  }
  float cre = 0.f, cim = 0.f;
  for (int c = 0; c < NCHUNK; ++c) {
    int ei = (b * NCHUNK + c) * STATE_F + n;
    cr_re[ei] = cre; cr_im[ei] = cim;
    float ere = el_re[ei], eim = el_im[ei];
    float nre = fmaf(pre, cre, fmaf(-pim, cim, ere));
    float nim = fmaf(pre, cim, fmaf(pim, cre, eim));
    cre = nre; cim = nim;
  }
}

// ---------------------------------------------------------------------------
// Kernel 2c: fix-up: states[j] += Lambda^{j+1} * carry(chunk), in place.
// ---------------------------------------------------------------------------
__global__ __launch_bounds__(256) void k_fixup(
    float* __restrict__ bu_re, float* __restrict__ bu_im,
    const float* __restrict__ nu_log, const float* __restrict__ theta_log,
    const float* __restrict__ cr_re, const float* __restrict__ cr_im) {
  const int c = blockIdx.x & (NCHUNK - 1);
  const int b = blockIdx.x >> 5;
  const int n = threadIdx.x;
  float lre, lim;
  lambda_of(n, nu_log, theta_log, lre, lim);
  int ei = (b * NCHUNK + c) * STATE_F + n;
  float cre = cr_re[ei], cim = cr_im[ei];
  float mre = lre * cre - lim * cim;   // Lambda * carry
  float mim = lre * cim + lim * cre;
  size_t base = ((size_t)b * TLEN + (size_t)c * CHUNK) * STATE_F + n;
  for (int j = 0; j < CHUNK; ++j) {
    size_t idx = base + (size_t)j * STATE_F;
    if (j + 16 < CHUNK) {
      __builtin_prefetch(&bu_re[idx + 16 * STATE_F], 0, 1);
      __builtin_prefetch(&bu_im[idx + 16 * STATE_F], 0, 1);
    }
    bu_re[idx] += mre;
    bu_im[idx] += mim;
    float nre = lre * mre - lim * mim;
    float nim = lre * mim + lim * mre;
    mre = nre; mim = nim;
  }
}

// ---------------------------------------------------------------------------
// Kernel 3: out = s_re @ C_re^T - s_im @ C_im^T + x @ D^T.
// WG = (otile, 8 mtiles). C_re / -C_im / D staged in LDS, shared by 8 waves.
// x loads + out stores nontemporal: keep Bu resident in the 192MB L2.
// ---------------------------------------------------------------------------
__global__ __launch_bounds__(256) void k_out_gemm(
    const float* __restrict__ s_re, const float* __restrict__ s_im,
    const float* __restrict__ x, const float* __restrict__ Cre,
    const float* __restrict__ Cim, const float* __restrict__ D,
    float* __restrict__ out) {
  __shared__ float sCre[16][PADC];
  __shared__ float sCim[16][PADC];   // holds -C_im
  __shared__ float sD[16][PADB];
  const int tid   = threadIdx.x;
  const int otile = blockIdx.x & 7;        // OUT_F/16 = 8
  const int mgrp  = blockIdx.x >> 3;       // 512 groups of 8 mtiles
  const int o0    = otile << 4;

  for (int q = tid; q < 1024; q += 256) {  // C_re: 16x256 = 1024 float4
    const int r = q >> 6, kq = (q & 63) << 2;
    const float4 v = *(const float4*)(Cre + (size_t)(o0 + r) * STATE_F + kq);
    float* dst = &sCre[r][kq];
    dst[0] = v.x; dst[1] = v.y; dst[2] = v.z; dst[3] = v.w;
  }
  for (int q = tid; q < 1024; q += 256) {  // -C_im
    const int r = q >> 6, kq = (q & 63) << 2;
    const float4 v = *(const float4*)(Cim + (size_t)(o0 + r) * STATE_F + kq);
    float* dst = &sCim[r][kq];
    dst[0] = -v.x; dst[1] = -v.y; dst[2] = -v.z; dst[3] = -v.w;
  }
  for (int q = tid; q < 512; q += 256) {   // D: 16x128 = 512 float4
    const int r = q >> 5, kq = (q & 31) << 2;
    const float4 v = *(const float4*)(D + (size_t)(o0 + r) * IN_F + kq);
    float* dst = &sD[r][kq];
    dst[0] = v.x; dst[1] = v.y; dst[2] = v.z; dst[3] = v.w;
  }
  __syncthreads();

  const int lane = tid & 31, wave = tid >> 5;
  const int m0   = (mgrp * 8 + wave) << 4;
  const int row  = lane & 15, khalf = lane >> 4;

  v8f acc = {};
  {  // states_re @ C_re^T
    const float* arow = s_re + (size_t)(m0 + row) * STATE_F + khalf * 2;
    const float* brow = &sCre[row][khalf * 2];
#pragma unroll 8
    for (int k0 = 0; k0 < STATE_F; k0 += 4)
      acc = wmma_f32_k4(*(const v2f*)(arow + k0), *(const v2f*)(brow + k0), acc);
  }
  {  // states_im @ (-C_im)^T  (negation pre-folded in LDS)
    const float* arow = s_im + (size_t)(m0 + row) * STATE_F + khalf * 2;
    const float* brow = &sCim[row][khalf * 2];
#pragma unroll 8
    for (int k0 = 0; k0 < STATE_F; k0 += 4)
      acc = wmma_f32_k4(*(const v2f*)(arow + k0), *(const v2f*)(brow + k0), acc);
  }
  {  // x @ D^T, x is last-use -> nontemporal
    const float* arow = x + (size_t)(m0 + row) * IN_F + khalf * 2;
    const float* brow = &sD[row][khalf * 2];
#pragma unroll 8
    for (int k0 = 0; k0 < IN_F; k0 += 4) {
      v2f a = __builtin_nontemporal_load((const v2f*)(arow + k0));
      acc = wmma_f32_k4(a, *(const v2f*)(brow + k0), acc);
    }
  }
  float* orow = out + o0 + row;
#pragma unroll
  for (int r = 0; r < 8; ++r)
    __builtin_nontemporal_store(acc[r],
                                &orow[(size_t)(m0 + r + khalf * 8) * OUT_F]);
}

// ---------------------------------------------------------------------------
extern "C" void kernel_launch(void* const* d_in, const int* in_sizes, int n_in,
                              void* d_out, int out_size, void* d_ws,
                              size_t ws_size, hipStream_t stream) {
  const float* x         = (const float*)d_in[0];
  const float* nu_log    = (const float*)d_in[1];
  const float* theta_log = (const float*)d_in[2];
  const float* gamma_log = (const float*)d_in[3];
  const float* B_re      = (const float*)d_in[4];
  const float* B_im      = (const float*)d_in[5];
  const float* C_re      = (const float*)d_in[6];
  const float* C_im      = (const float*)d_in[7];
  const float* D         = (const float*)d_in[8];
  float* out = (float*)d_out;

  float* ws = (float*)d_ws;
  const size_t MN = (size_t)MTOT * STATE_F;            // 16.7M floats
  const size_t ES = (size_t)BATCH * NCHUNK * STATE_F;  // 65536 floats
  float* bu_re = ws;
  float* bu_im = ws + MN;
  float* el_re = ws + 2 * MN;
  float* el_im = el_re + ES;
  float* cr_re = el_im + ES;
  float* cr_im = cr_re + ES;

  dim3 blk(256);
  // 1) Bu GEMMs: 16 ntiles x 512 mtile-groups
  k_bu_gemm<<<(STATE_F / 16) * (MTOT / 16 / 8), blk, 0, stream>>>(
      x, B_re, B_im, gamma_log, bu_re, bu_im);
  // 2) chunked scan
  k_scan_local<<<BATCH * NCHUNK, blk, 0, stream>>>(bu_re, bu_im, nu_log,
                                                   theta_log, el_re, el_im);
  k_carry<<<BATCH * STATE_F / 256, blk, 0, stream>>>(nu_log, theta_log, el_re,
                                                     el_im, cr_re, cr_im);
  k_fixup<<<BATCH * NCHUNK, blk, 0, stream>>>(bu_re, bu_im, nu_log, theta_log,
                                              cr_re, cr_im);
  // 3) output GEMM: 8 otiles x 512 mtile-groups
  k_out_gemm<<<(OUT_F / 16) * (MTOT / 16 / 8), blk, 0, stream>>>(
      bu_re, bu_im, x, C_re, C_im, D, out);
}